// xsphModule_47021301957215
// MI455X (gfx1250) — compile-verified
//
#include <hip/hip_runtime.h>

// XSPH viscosity correction (SPH edge scatter), CDNA5 / gfx1250.
//
// Bottleneck analysis: E=8.4M edges stream 12B each (i,j,q) from HBM, but the
// per-edge random gathers dominate L2 transaction throughput. So:
//  (1) precompute packed 16B particle records {rho, fca, vx, vy} in d_ws
//      -> edge pass does only TWO b128 gathers + 2 FP atomics per edge;
//  (2) stream the edge data through the CDNA5 async global->LDS path
//      (GLOBAL_LOAD_ASYNC_TO_LDS_B32, ASYNCcnt) with per-wave double
//      buffering so tile t+1 is in flight while tile t computes;
//  (3) scatter with native global_atomic_add_f32 (unsafeAtomicAdd).

#define XSPH_PI 3.14159265358979323846f
#define BLOCK 256
#define WAVES_PER_BLOCK (BLOCK / 32)
#define TILES_PER_WAVE 8

#if defined(__has_builtin)
#if __has_builtin(__builtin_amdgcn_global_load_async_to_lds_b32)
#define HAVE_ASYNC_BUILTIN 1
#endif
#if __has_builtin(__builtin_amdgcn_s_wait_asynccnt)
#define HAVE_WAIT_ASYNC_BUILTIN 1
#endif
#endif

// Builtin signature (probe-confirmed by hipcc diagnostic): (global int*, shared int*, imm, imm)
typedef __attribute__((address_space(1))) int g_i32_t;
typedef __attribute__((address_space(3))) int lds_i32_t;

__device__ __forceinline__ void async_copy_b32(const void* gptr, void* lptr) {
#if defined(HAVE_ASYNC_BUILTIN)
    __builtin_amdgcn_global_load_async_to_lds_b32(
        (g_i32_t*)(unsigned long long)gptr,
        (lds_i32_t*)(unsigned int)(unsigned long long)lptr,
        0, 0);
#else
    unsigned int lds_addr = (unsigned int)(unsigned long long)lptr;
    unsigned long long gaddr = (unsigned long long)gptr;
    asm volatile("global_load_async_to_lds_b32 %0, %1, off"
                 :
                 : "v"(lds_addr), "v"(gaddr)
                 : "memory");
#endif
}

template <int N>
__device__ __forceinline__ void wait_asynccnt() {
#if defined(HAVE_WAIT_ASYNC_BUILTIN)
    __builtin_amdgcn_s_wait_asynccnt(N);
    asm volatile("" ::: "memory");  // keep LDS reads after the wait
#else
    asm volatile("s_wait_asynccnt %0" :: "i"(N) : "memory");
#endif
}

__global__ __launch_bounds__(BLOCK) void xsph_zero_kernel(float* __restrict__ out, int n) {
    int t = (blockIdx.x * blockDim.x + threadIdx.x) * 4;
    if (t + 3 < n) {
        *reinterpret_cast<float4*>(out + t) = float4{0.f, 0.f, 0.f, 0.f};
    } else {
        int lim = n < t + 4 ? n : t + 4;
        for (int k = t; k < lim; ++k) out[k] = 0.f;
    }
}

// Pack per-particle record: {rho = density*restDensity, fca = fc*restDensity*area, vx, vy}
__global__ __launch_bounds__(BLOCK) void xsph_pack_kernel(
    const float*  __restrict__ fluidCoefficient,
    const float*  __restrict__ area,
    const float*  __restrict__ density,
    const float*  __restrict__ restDensity,
    const float2* __restrict__ velocity,
    float4*       __restrict__ packed,
    int n)
{
    int t = blockIdx.x * blockDim.x + threadIdx.x;
    if (t >= n) return;
    const float fc = fluidCoefficient[0];
    const float rd = restDensity[t];
    const float2 v = velocity[t];
    packed[t] = float4{density[t] * rd, fc * rd * area[t], v.x, v.y};
}

// Fast path: two b128 gathers per edge from the packed particle table.
__global__ __launch_bounds__(BLOCK) void xsph_edge_packed_kernel(
    const float4* __restrict__ packed,            // [N] {rho, fca, vx, vy}
    const float*  __restrict__ radial,            // [E]
    const int*    __restrict__ nbrI,              // [E]
    const int*    __restrict__ nbrJ,              // [E]
    const float*  __restrict__ support,           // [1]
    float*        __restrict__ out,               // [N*2]
    int nEdges)
{
    __shared__ unsigned int stage[WAVES_PER_BLOCK][2][3][32];

    const float h = support[0];
    const float C = 7.0f / (XSPH_PI * h * h);

    const int lane   = threadIdx.x & 31;
    const int wv     = threadIdx.x >> 5;
    const int gwave  = blockIdx.x * WAVES_PER_BLOCK + wv;
    const int stride = gridDim.x * BLOCK;
    const int eLast  = nEdges - 1;

    int e = gwave * 32 + lane;

    {   // Prologue: stage tile 0 (clamped addresses for OOB lanes).
        int el = (e > eLast) ? eLast : e;
        async_copy_b32(&nbrI[el],   &stage[wv][0][0][lane]);
        async_copy_b32(&nbrJ[el],   &stage[wv][0][1][lane]);
        async_copy_b32(&radial[el], &stage[wv][0][2][lane]);
    }

    int buf = 0;
    for (; (e - lane) <= eLast; e += stride, buf ^= 1) {  // wave-uniform bound
        const int  enext = e + stride;
        const bool more  = (enext - lane) <= eLast;       // wave-uniform

        if (more) {
            int el = (enext > eLast) ? eLast : enext;
            async_copy_b32(&nbrI[el],   &stage[wv][buf ^ 1][0][lane]);
            async_copy_b32(&nbrJ[el],   &stage[wv][buf ^ 1][1][lane]);
            async_copy_b32(&radial[el], &stage[wv][buf ^ 1][2][lane]);
            wait_asynccnt<3>();   // tile t landed; t+1 in flight
        } else {
            wait_asynccnt<0>();   // drain for last tile
        }

        const int   i = (int)stage[wv][buf][0][lane];
        const int   j = (int)stage[wv][buf][1][lane];
        const float q = __uint_as_float(stage[wv][buf][2][lane]);

        if (e <= eLast) {
            const float4 pi = packed[i];   // one 16B L2 transaction
            const float4 pj = packed[j];   // one 16B L2 transaction

            const float b  = fmaxf(1.0f - q, 0.0f);
            const float b2 = b * b;
            const float w  = C * (b2 * b2) * fmaf(4.0f, q, 1.0f);
            const float s  = (2.0f * pj.y * w) / (pi.x + pj.x);

            float* dst = out + 2 * i;
            unsafeAtomicAdd(dst,     s * (pj.z - pi.z));
            unsafeAtomicAdd(dst + 1, s * (pj.w - pi.w));
        }
    }
}

// Fallback (no workspace): direct gathers, same async-staged edge stream.
__global__ __launch_bounds__(BLOCK) void xsph_edge_direct_kernel(
    const float*  __restrict__ fluidCoefficient,
    const float*  __restrict__ area,
    const float*  __restrict__ density,
    const float*  __restrict__ restDensity,
    const float2* __restrict__ velocity,
    const float*  __restrict__ radial,
    const int*    __restrict__ nbrI,
    const int*    __restrict__ nbrJ,
    const float*  __restrict__ support,
    float*        __restrict__ out,
    int nEdges)
{
    __shared__ unsigned int stage[WAVES_PER_BLOCK][2][3][32];

    const float fc = fluidCoefficient[0];
    const float h  = support[0];
    const float C  = 7.0f / (XSPH_PI * h * h);

    const int lane   = threadIdx.x & 31;
    const int wv     = threadIdx.x >> 5;
    const int gwave  = blockIdx.x * WAVES_PER_BLOCK + wv;
    const int stride = gridDim.x * BLOCK;
    const int eLast  = nEdges - 1;

    int e = gwave * 32 + lane;
    {
        int el = (e > eLast) ? eLast : e;
        async_copy_b32(&nbrI[el],   &stage[wv][0][0][lane]);
        async_copy_b32(&nbrJ[el],   &stage[wv][0][1][lane]);
        async_copy_b32(&radial[el], &stage[wv][0][2][lane]);
    }
    int buf = 0;
    for (; (e - lane) <= eLast; e += stride, buf ^= 1) {
        const int  enext = e + stride;
        const bool more  = (enext - lane) <= eLast;
        if (more) {
            int el = (enext > eLast) ? eLast : enext;
            async_copy_b32(&nbrI[el],   &stage[wv][buf ^ 1][0][lane]);
            async_copy_b32(&nbrJ[el],   &stage[wv][buf ^ 1][1][lane]);
            async_copy_b32(&radial[el], &stage[wv][buf ^ 1][2][lane]);
            wait_asynccnt<3>();
        } else {
            wait_asynccnt<0>();
        }
        const int   i = (int)stage[wv][buf][0][lane];
        const int   j = (int)stage[wv][buf][1][lane];
        const float q = __uint_as_float(stage[wv][buf][2][lane]);
        if (e <= eLast) {
            const float  rdi = restDensity[i];
            const float  rdj = restDensity[j];
            const float  fac = fc * rdj * area[j];
            const float  rho = fmaf(density[i], rdi, density[j] * rdj);
            const float2 vi  = velocity[i];
            const float2 vj  = velocity[j];
            const float b  = fmaxf(1.0f - q, 0.0f);
            const float b2 = b * b;
            const float w  = C * (b2 * b2) * fmaf(4.0f, q, 1.0f);
            const float s  = (2.0f * fac * w) / rho;
            float* dst = out + 2 * i;
            unsafeAtomicAdd(dst,     s * (vj.x - vi.x));
            unsafeAtomicAdd(dst + 1, s * (vj.y - vi.y));
        }
    }
}

extern "C" void kernel_launch(void* const* d_in, const int* in_sizes, int n_in,
                              void* d_out, int out_size, void* d_ws, size_t ws_size,
                              hipStream_t stream) {
    // setup_inputs() order:
    // 0 fluidCoefficient [1] f32, 1 area [N], 2 density [N], 3 restDensity [N],
    // 4 velocity [N,2], 5 radialDistance [E], 6 neighbors [2,E] i32, 7 support [1]
    const float*  fc      = (const float*)d_in[0];
    const float*  area    = (const float*)d_in[1];
    const float*  den     = (const float*)d_in[2];
    const float*  rd      = (const float*)d_in[3];
    const float2* vel     = (const float2*)d_in[4];
    const float*  radial  = (const float*)d_in[5];
    const int*    nbr     = (const int*)d_in[6];
    const float*  support = (const float*)d_in[7];
    float*        out     = (float*)d_out;

    const int nParticles = in_sizes[1];
    const int nEdges     = in_sizes[5];
    const int* nbrI      = nbr;
    const int* nbrJ      = nbr + nEdges;

    // Zero output (harness poisons d_out).
    {
        int threads = (out_size + 3) / 4;
        int blocks  = (threads + BLOCK - 1) / BLOCK;
        if (blocks < 1) blocks = 1;
        xsph_zero_kernel<<<blocks, BLOCK, 0, stream>>>(out, out_size);
    }

    const int edgesPerWave = 32 * TILES_PER_WAVE;
    int totalWaves = (nEdges + edgesPerWave - 1) / edgesPerWave;
    int eBlocks    = (totalWaves + WAVES_PER_BLOCK - 1) / WAVES_PER_BLOCK;
    if (eBlocks < 1) eBlocks = 1;

    const size_t packedBytes = (size_t)nParticles * sizeof(float4);
    if (ws_size >= packedBytes) {
        // Pack per-particle records, then 2-gather edge pass.
        float4* packed = (float4*)d_ws;
        int pBlocks = (nParticles + BLOCK - 1) / BLOCK;
        if (pBlocks < 1) pBlocks = 1;
        xsph_pack_kernel<<<pBlocks, BLOCK, 0, stream>>>(
            fc, area, den, rd, vel, packed, nParticles);
        xsph_edge_packed_kernel<<<eBlocks, BLOCK, 0, stream>>>(
            packed, radial, nbrI, nbrJ, support, out, nEdges);
    } else {
        xsph_edge_direct_kernel<<<eBlocks, BLOCK, 0, stream>>>(
            fc, area, den, rd, vel, radial, nbrI, nbrJ, support, out, nEdges);
    }
}